// GraphSAGEFraudDetector_63917703299119
// MI455X (gfx1250) — compile-verified
//
#include <hip/hip_runtime.h>
#include <math.h>

#define N_NODES 50000
#define N_EDGES 640000

typedef __attribute__((ext_vector_type(2))) float v2f;
typedef __attribute__((ext_vector_type(8))) float v8f;

// ---------------------------------------------------------------------------
// utility: zero-fill
// ---------------------------------------------------------------------------
__global__ void zero_f32(float* __restrict__ p, long n) {
  long i = (long)blockIdx.x * blockDim.x + threadIdx.x;
  long stride = (long)gridDim.x * blockDim.x;
  for (; i < n; i += stride) p[i] = 0.0f;
}

// ---------------------------------------------------------------------------
// degree: deg[dst[e]] += 1
// ---------------------------------------------------------------------------
__global__ void degree_kernel(const long long* __restrict__ dst,
                              float* __restrict__ deg, int E) {
  int e = blockIdx.x * blockDim.x + threadIdx.x;
  if (e < E) {
    int d = (int)dst[e];
    __hip_atomic_fetch_add(&deg[d], 1.0f, __ATOMIC_RELAXED,
                           __HIP_MEMORY_SCOPE_AGENT);
  }
}

// deg -> 1 / max(deg, 1)   (in place; reused by all three layers)
__global__ void invdeg_kernel(float* __restrict__ deg, int n) {
  int i = blockIdx.x * blockDim.x + threadIdx.x;
  if (i < n) deg[i] = 1.0f / fmaxf(deg[i], 1.0f);
}

// ---------------------------------------------------------------------------
// scatter-mean: agg[dst] += h[src] * invdeg[dst]
// thread = (edge, group-of-4 channels); float4 gather, 4 f32 atomics
// ---------------------------------------------------------------------------
__global__ void scatter_mean_kernel(const float* __restrict__ h,
                                    const long long* __restrict__ src,
                                    const long long* __restrict__ dst,
                                    const float* __restrict__ invdeg,
                                    float* __restrict__ agg, int C) {
  int per_edge = C >> 2;                       // float4 groups per edge
  long t = (long)blockIdx.x * blockDim.x + threadIdx.x;
  long total = (long)N_EDGES * per_edge;
  if (t >= total) return;
  int e  = (int)(t / per_edge);
  int c4 = (int)(t - (long)e * per_edge);
  int s = (int)src[e];
  int d = (int)dst[e];
  const float4 v = *(const float4*)(h + (long)s * C + c4 * 4);
  float w = invdeg[d];
  float* o = agg + (long)d * C + c4 * 4;
  __hip_atomic_fetch_add(o + 0, v.x * w, __ATOMIC_RELAXED, __HIP_MEMORY_SCOPE_AGENT);
  __hip_atomic_fetch_add(o + 1, v.y * w, __ATOMIC_RELAXED, __HIP_MEMORY_SCOPE_AGENT);
  __hip_atomic_fetch_add(o + 2, v.z * w, __ATOMIC_RELAXED, __HIP_MEMORY_SCOPE_AGENT);
  __hip_atomic_fetch_add(o + 3, v.w * w, __ATOMIC_RELAXED, __HIP_MEMORY_SCOPE_AGENT);
}

// ---------------------------------------------------------------------------
// SAGE dense phase, exact fp32 via V_WMMA_F32_16X16X4_F32:
//   out = agg @ Wl + h @ Wr + b  (optional ReLU)
// One wave32 computes one 16x16 output tile. N_NODES % 16 == 0, so no row
// predication; the only branch (tile overflow) is wave-uniform -> EXEC stays
// all ones as WMMA requires.
//
// VGPR layouts (ISA 7.12.2, 32-bit):
//   A 16x4 : lanes 0-15 -> M=lane, K={k,k+1}; lanes 16-31 -> K={k+2,k+3}
//   B 4x16 : lanes 0-15 -> N=lane, rows K={k,k+1}; lanes 16-31 rows {k+2,k+3}
//   C/D    : n = lane&15, m = vgpr + 8*(lane>>4)
// ---------------------------------------------------------------------------
__global__ void sage_gemm_wmma(const float* __restrict__ agg,
                               const float* __restrict__ h,
                               const float* __restrict__ Wl,
                               const float* __restrict__ Wr,
                               const float* __restrict__ bias,
                               float* __restrict__ out,
                               int Cin, int Cout, int relu, int numTiles) {
  const int wave = threadIdx.x >> 5;
  const int lane = threadIdx.x & 31;
  const int tile = blockIdx.x * (blockDim.x >> 5) + wave;
  if (tile >= numTiles) return;                 // wave-uniform

  const int colTiles = Cout >> 4;
  const int tm = tile / colTiles;               // 16-node row tile
  const int tn = tile - tm * colTiles;          // 16-channel col tile

  const int m  = lane & 15;
  const int g  = lane >> 4;                     // half-wave selects K pair
  const long rowBase = ((long)tm * 16 + m) * Cin + 2 * g;
  const int n0 = tn * 16 + (lane & 15);

  v8f acc = {};

  // agg @ Wl
  for (int k = 0; k < Cin; k += 4) {
    v2f a = *(const v2f*)(agg + rowBase + k);
    v2f b;
    b.x = Wl[(k + 2 * g) * Cout + n0];
    b.y = Wl[(k + 2 * g + 1) * Cout + n0];
    acc = __builtin_amdgcn_wmma_f32_16x16x4_f32(
        /*neg_a=*/false, a, /*neg_b=*/false, b,
        /*c_mod=*/(short)0, acc, /*reuse_a=*/false, /*reuse_b=*/false);
  }
  // h @ Wr (accumulate into same tile)
  for (int k = 0; k < Cin; k += 4) {
    v2f a = *(const v2f*)(h + rowBase + k);
    v2f b;
    b.x = Wr[(k + 2 * g) * Cout + n0];
    b.y = Wr[(k + 2 * g + 1) * Cout + n0];
    acc = __builtin_amdgcn_wmma_f32_16x16x4_f32(
        false, a, false, b, (short)0, acc, false, false);
  }

  const float bv = bias[n0];
#pragma unroll
  for (int r = 0; r < 8; ++r) {
    int mr = r + 8 * g;                         // D-matrix row for this VGPR
    float v = acc[r] + bv;
    if (relu) v = fmaxf(v, 0.0f);
    out[((long)tm * 16 + mr) * Cout + n0] = v;
  }
}

// ---------------------------------------------------------------------------
// classifier: sigmoid(relu(h@Wc1 + bc1) @ Wc2 + bc2); Wc1 staged in LDS
// ---------------------------------------------------------------------------
__global__ void classifier_kernel(const float* __restrict__ h,
                                  const float* __restrict__ Wc1,
                                  const float* __restrict__ bc1,
                                  const float* __restrict__ Wc2,
                                  const float* __restrict__ bc2,
                                  float* __restrict__ out) {
  __shared__ float sW1[64 * 32];
  __shared__ float sb1[32];
  __shared__ float sW2[32];
  __shared__ float sb2;
  for (int i = threadIdx.x; i < 64 * 32; i += blockDim.x) sW1[i] = Wc1[i];
  if (threadIdx.x < 32) {
    sb1[threadIdx.x] = bc1[threadIdx.x];
    sW2[threadIdx.x] = Wc2[threadIdx.x];
  }
  if (threadIdx.x == 0) sb2 = bc2[0];
  __syncthreads();

  int node = blockIdx.x * blockDim.x + threadIdx.x;
  if (node >= N_NODES) return;

  float xr[64];
  const float4* hp = (const float4*)(h + (long)node * 64);
#pragma unroll
  for (int i = 0; i < 16; ++i) {
    float4 v = hp[i];
    xr[4 * i + 0] = v.x; xr[4 * i + 1] = v.y;
    xr[4 * i + 2] = v.z; xr[4 * i + 3] = v.w;
  }
  float z = sb2;
  for (int j = 0; j < 32; ++j) {
    float t = sb1[j];
#pragma unroll
    for (int i = 0; i < 64; ++i) t = fmaf(xr[i], sW1[i * 32 + j], t);
    t = fmaxf(t, 0.0f);
    z = fmaf(t, sW2[j], z);
  }
  out[node] = 1.0f / (1.0f + expf(-z));
}

// ---------------------------------------------------------------------------
// launch
// ---------------------------------------------------------------------------
extern "C" void kernel_launch(void* const* d_in, const int* in_sizes, int n_in,
                              void* d_out, int out_size, void* d_ws,
                              size_t ws_size, hipStream_t stream) {
  const float*     x   = (const float*)d_in[0];
  const long long* ei  = (const long long*)d_in[1];   // jnp.int64 per reference
  const long long* src = ei;                          // edge_index[0]
  const long long* dst = ei + N_EDGES;                // edge_index[1]
  const float* Wl0 = (const float*)d_in[2];
  const float* Wr0 = (const float*)d_in[3];
  const float* b0  = (const float*)d_in[4];
  const float* Wl1 = (const float*)d_in[5];
  const float* Wr1 = (const float*)d_in[6];
  const float* b1  = (const float*)d_in[7];
  const float* Wl2 = (const float*)d_in[8];
  const float* Wr2 = (const float*)d_in[9];
  const float* b2  = (const float*)d_in[10];
  const float* Wc1 = (const float*)d_in[11];
  const float* bc1 = (const float*)d_in[12];
  const float* Wc2 = (const float*)d_in[13];
  const float* bc2 = (const float*)d_in[14];
  float* out = (float*)d_out;

  // workspace layout (floats): deg[N] | agg[N*128] | hA[N*128] | hB[N*128]
  float* deg = (float*)d_ws;
  float* agg = deg + N_NODES;
  float* hA  = agg + (long)N_NODES * 128;
  float* hB  = hA  + (long)N_NODES * 128;

  const int T = 256;

  // degree -> inv-degree (constant across layers)
  zero_f32<<<256, T, 0, stream>>>(deg, (long)N_NODES);
  degree_kernel<<<(N_EDGES + T - 1) / T, T, 0, stream>>>(dst, deg, N_EDGES);
  invdeg_kernel<<<(N_NODES + T - 1) / T, T, 0, stream>>>(deg, N_NODES);

  // ---- layer 0: Cin=64 -> Cout=128, ReLU ----
  zero_f32<<<2048, T, 0, stream>>>(agg, (long)N_NODES * 64);
  {
    long tot = (long)N_EDGES * (64 / 4);
    scatter_mean_kernel<<<(int)((tot + T - 1) / T), T, 0, stream>>>(
        x, src, dst, deg, agg, 64);
    int tiles = (N_NODES / 16) * (128 / 16);
    sage_gemm_wmma<<<(tiles + 7) / 8, T, 0, stream>>>(
        agg, x, Wl0, Wr0, b0, hA, 64, 128, 1, tiles);
  }

  // ---- layer 1: Cin=128 -> Cout=128, ReLU ----
  zero_f32<<<2048, T, 0, stream>>>(agg, (long)N_NODES * 128);
  {
    long tot = (long)N_EDGES * (128 / 4);
    scatter_mean_kernel<<<(int)((tot + T - 1) / T), T, 0, stream>>>(
        hA, src, dst, deg, agg, 128);
    int tiles = (N_NODES / 16) * (128 / 16);
    sage_gemm_wmma<<<(tiles + 7) / 8, T, 0, stream>>>(
        agg, hA, Wl1, Wr1, b1, hB, 128, 128, 1, tiles);
  }

  // ---- layer 2: Cin=128 -> Cout=64, no ReLU ----
  zero_f32<<<2048, T, 0, stream>>>(agg, (long)N_NODES * 128);
  {
    long tot = (long)N_EDGES * (128 / 4);
    scatter_mean_kernel<<<(int)((tot + T - 1) / T), T, 0, stream>>>(
        hB, src, dst, deg, agg, 128);
    int tiles = (N_NODES / 16) * (64 / 16);
    sage_gemm_wmma<<<(tiles + 7) / 8, T, 0, stream>>>(
        agg, hB, Wl2, Wr2, b2, hA, 128, 64, 0, tiles);
  }

  // ---- classifier ----
  classifier_kernel<<<(N_NODES + T - 1) / T, T, 0, stream>>>(
      hA, Wc1, bc1, Wc2, bc2, out);
}